// TopkPointExtractor_40656160424616
// MI455X (gfx1250) — compile-verified
//
#include <hip/hip_runtime.h>
#include <hip/hip_bf16.h>
#include <math.h>

typedef __attribute__((ext_vector_type(2))) float v2f;
typedef __attribute__((ext_vector_type(8))) float v8f;

#define Bd   16
#define Cd   256
#define Hd   56
#define Wd   56
#define HW   (Hd * Wd)      // 3136
#define KTOP 256
#define NSORT 4096          // next pow2 >= HW

// ---------------------------------------------------------------------------
// Kernel A: per (b, y) compute norm[x], dotH[x], dotV[x] via fp32 WMMA Gram
// tiles. 4 waves per block; wave w owns pixel-tile start s in {0,14,28,42}.
//   HT = Gram(row y, row y)        -> diag = squared norms, (m, m+2) = dotH
//   VT = Gram(row y-1, row y+1)    -> diag = dotV
// ---------------------------------------------------------------------------
__global__ __launch_bounds__(128) void sim_parts_kernel(
    const float* __restrict__ F,
    float* __restrict__ normOut,
    float* __restrict__ dotHOut,
    float* __restrict__ dotVOut)
{
    __shared__ float HT[4][16][16];
    __shared__ float VT[4][16][16];

    const int y    = blockIdx.x;
    const int b    = blockIdx.y;
    const int tid  = threadIdx.x;
    const int wave = tid >> 5;
    const int lane = tid & 31;

    const int s  = wave * 14;                    // 0,14,28,42
    const int ym = (y > 0)      ? y - 1 : 0;
    const int yp = (y < Hd - 1) ? y + 1 : Hd - 1;

    int p = s + (lane & 15);
    if (p > Wd - 1) p = Wd - 1;                  // replicate-clamp (dup cols unused)
    const int khalf = (lane >> 4) << 1;          // 0 or 2: K pair per lane half

    const size_t cs = (size_t)HW;                // channel stride in elements
    const float* baseH  = F + (size_t)b * Cd * cs + (size_t)y  * Wd + p;
    const float* baseVm = F + (size_t)b * Cd * cs + (size_t)ym * Wd + p;
    const float* baseVp = F + (size_t)b * Cd * cs + (size_t)yp * Wd + p;

    v8f cH = {};
    v8f cV = {};
    for (int c0 = 0; c0 < Cd; c0 += 4) {
        const size_t o0 = (size_t)(c0 + khalf) * cs;
        const size_t o1 = o0 + cs;
        v2f aH; aH.x = baseH[o0];  aH.y = baseH[o1];
        v2f aV; aV.x = baseVm[o0]; aV.y = baseVm[o1];
        v2f bV; bV.x = baseVp[o0]; bV.y = baseVp[o1];
        // D = A(16x4) * B(4x16) + C, fp32 exact
        cH = __builtin_amdgcn_wmma_f32_16x16x4_f32(false, aH, false, aH, (short)0, cH, false, false);
        cV = __builtin_amdgcn_wmma_f32_16x16x4_f32(false, aV, false, bV, (short)0, cV, false, false);
    }

    // Spill C frags: lane l, VGPR v holds G[v + 8*(l>=16)][l & 15]
    const int rbase = (lane >> 4) * 8;
    const int col   = lane & 15;
#pragma unroll
    for (int v = 0; v < 8; ++v) {
        HT[wave][rbase + v][col] = cH[v];
        VT[wave][rbase + v][col] = cV[v];
    }
    __syncthreads();

    if (tid < Wd) {
        const int x  = tid;
        const int tN = (x < 14) ? 0 : (x < 28) ? 1 : (x < 42) ? 2 : 3;
        const int mN = x - tN * 14;
        const float nrm = sqrtf(HT[tN][mN][mN]);
        const float dv  = VT[tN][mN][mN];
        float dh;
        if (x == 0)            dh = HT[0][0][1];          // pair (0,1)
        else if (x == Wd - 1)  dh = HT[3][12][13];        // pair (54,55)
        else {
            const int tH = (x <= 14) ? 0 : (x <= 28) ? 1 : (x <= 42) ? 2 : 3;
            const int m  = x - 1 - tH * 14;
            dh = HT[tH][m][m + 2];                        // pair (x-1, x+1)
        }
        const size_t o = ((size_t)b * Hd + y) * Wd + x;
        normOut[o] = nrm;
        dotHOut[o] = dh;
        dotVOut[o] = dv;
    }
}

// ---------------------------------------------------------------------------
// Kernel B: fused sim-combine + per-batch top-256.
//   Load phase computes sim[i] = 0.5*(dotH/(n[xl]n[xr]) + dotV/(n[ym]n[yp]))
//   directly into LDS (padded to 4096 with -inf), then a bitonic sort on
//   (val desc, idx asc) keys -- exact jax.lax.top_k ordering, 78 barriers
//   instead of ~2800 for iterative argmax.
// ---------------------------------------------------------------------------
__global__ __launch_bounds__(1024) void topk_kernel(
    const float* __restrict__ norm,
    const float* __restrict__ dotH,
    const float* __restrict__ dotV,
    float* __restrict__ valsOut,
    float* __restrict__ yOut,
    float* __restrict__ xOut,
    int*   __restrict__ idxOut)
{
    __shared__ float sval[NSORT];
    __shared__ int   sidx[NSORT];

    const int b   = blockIdx.x;
    const int tid = threadIdx.x;
    const float* nb = norm + (size_t)b * HW;

    // Fused combine + load
    for (int i = tid; i < NSORT; i += 1024) {
        float v = -INFINITY;
        if (i < HW) {
            const int y  = i / Wd;
            const int x  = i % Wd;
            const int xl = (x > 0)      ? x - 1 : 0;
            const int xr = (x < Wd - 1) ? x + 1 : Wd - 1;
            const int ym = (y > 0)      ? y - 1 : 0;
            const int yp = (y < Hd - 1) ? y + 1 : Hd - 1;
            const size_t g = (size_t)b * HW + i;
            const float ch = dotH[g] / (nb[y * Wd + xl] * nb[y * Wd + xr]);
            const float cv = dotV[g] / (nb[ym * Wd + x] * nb[yp * Wd + x]);
            v = 0.5f * (ch + cv);
        }
        sval[i] = v;
        sidx[i] = i;
    }
    __syncthreads();

    // Bitonic sort, best-first ("ascending" == better): val desc, idx asc
    for (int kk = 2; kk <= NSORT; kk <<= 1) {
        for (int jj = kk >> 1; jj > 0; jj >>= 1) {
            for (int i = tid; i < NSORT; i += 1024) {
                const int ixj = i ^ jj;
                if (ixj > i) {
                    const float v0 = sval[i],   v1 = sval[ixj];
                    const int   i0 = sidx[i],   i1 = sidx[ixj];
                    const bool firstBetter = (v0 > v1) || (v0 == v1 && i0 < i1);
                    const bool up = ((i & kk) == 0);
                    if (up != firstBetter) {
                        sval[i] = v1; sval[ixj] = v0;
                        sidx[i] = i1; sidx[ixj] = i0;
                    }
                }
            }
            __syncthreads();
        }
    }

    if (tid < KTOP) {
        const int p = sidx[tid];
        valsOut[b * KTOP + tid] = sval[tid];
        yOut[b * KTOP + tid]    = (float)(p / Wd);
        xOut[b * KTOP + tid]    = (float)(p % Wd);
        idxOut[b * KTOP + tid]  = p;
    }
}

// ---------------------------------------------------------------------------
// Kernel C: gather point features -> out[b][k][c] = F[b][c][p]
// ---------------------------------------------------------------------------
__global__ __launch_bounds__(256) void gather_kernel(
    const float* __restrict__ F,
    const int* __restrict__ idx,
    float* __restrict__ out)
{
    const int bk = blockIdx.x;          // b*KTOP + k
    const int b  = bk >> 8;
    const int c  = threadIdx.x;
    const int p  = idx[bk];
    out[(size_t)bk * Cd + c] = F[((size_t)b * Cd + c) * HW + p];
}

// ---------------------------------------------------------------------------
extern "C" void kernel_launch(void* const* d_in, const int* in_sizes, int n_in,
                              void* d_out, int out_size, void* d_ws, size_t ws_size,
                              hipStream_t stream)
{
    const float* F   = (const float*)d_in[0];
    float* out = (float*)d_out;

    float* wsf  = (float*)d_ws;
    float* norm = wsf;
    float* dH   = wsf + (size_t)Bd * HW;
    float* dV   = wsf + (size_t)2 * Bd * HW;
    int*   tidx = (int*)(wsf + (size_t)3 * Bd * HW);

    float* pf   = out;                                   // [B,K,C]
    float* vals = out + (size_t)Bd * KTOP * Cd;          // [B,K]
    float* yc   = vals + (size_t)Bd * KTOP;              // [B,K]
    float* xc   = yc   + (size_t)Bd * KTOP;              // [B,K]

    dim3 gA(Hd, Bd);
    sim_parts_kernel<<<gA, 128, 0, stream>>>(F, norm, dH, dV);

    topk_kernel<<<Bd, 1024, 0, stream>>>(norm, dH, dV, vals, yc, xc, tidx);

    gather_kernel<<<Bd * KTOP, Cd, 0, stream>>>(F, tidx, pf);
}